// DelayGIN_40604620817035
// MI455X (gfx1250) — compile-verified
//
#include <hip/hip_runtime.h>
#include <hip/hip_bf16.h>

#define DDIM 128
#define LDS_STRIDE 132   // 132 % 64 == 4 -> A-fragment LDS reads are bank-conflict-free

typedef __attribute__((ext_vector_type(2))) float v2f;
typedef __attribute__((ext_vector_type(8))) float v8f;

// -------------------------------------------------------------------------
// Scatter-add: one wave32 per edge. Relation test is wave-uniform (no
// divergence). Each lane gathers float4 (16B) of the 512B source row and
// atomically accumulates into agg[dst]. This is the roofline-bound phase
// (~1.5GB random gathers + ~3GB atomic RMW across the whole model).
// -------------------------------------------------------------------------
__global__ void __launch_bounds__(256)
scatter_add_kernel(const float* __restrict__ h,
                   const int* __restrict__ src,
                   const int* __restrict__ dst,
                   const int* __restrict__ attr,
                   int rel,
                   float* __restrict__ agg,
                   int E)
{
    const int lane = threadIdx.x & 31;
    const int e = blockIdx.x * (blockDim.x >> 5) + (threadIdx.x >> 5);
    if (e >= E) return;
    if (attr[e] != rel) return;

    const int s  = src[e];
    const int d0 = dst[e];

    const float4 v = ((const float4*)(h + (size_t)s * DDIM))[lane];
    float* ap = agg + (size_t)d0 * DDIM + lane * 4;
    atomicAdd(ap + 0, v.x);
    atomicAdd(ap + 1, v.y);
    atomicAdd(ap + 2, v.z);
    atomicAdd(ap + 3, v.w);
}

// -------------------------------------------------------------------------
// out[N,128] += relu(A[N,128] @ W[128,128] + bias) via V_WMMA_F32_16X16X4_F32.
// Block = 256 threads = 8 waves; block owns a 32-row stripe of A staged in
// LDS; wave w owns output columns [16w, 16w+16). Each wave runs two
// accumulator chains (rows m and m+16) sharing one B fragment per k-step,
// so every W load feeds 2 WMMAs. K=128 swept in 32 steps of K=4.
// Fragment layouts per CDNA5 ISA 7.12.2:
//   A 16x4 : lane m = lane&15, VGPR0 = K{kb}, VGPR1 = K{kb+1}, kb=(lane>>4)*2
//   B 4x16 : lane n = lane&15, same K striping as A
//   C 16x16: VGPR i -> row (lane>>4)*8 + i, col = lane&15
// -------------------------------------------------------------------------
__global__ void __launch_bounds__(256)
gemm32_relu_acc_kernel(const float* __restrict__ A,
                       const float* __restrict__ W,
                       const float* __restrict__ bias,
                       float* __restrict__ out,
                       int N)
{
    __shared__ float As[32 * LDS_STRIDE];

    const int tid  = threadIdx.x;
    const int lane = tid & 31;
    const int wave = tid >> 5;            // 0..7 -> column tile
    const int row0 = blockIdx.x * 32;

    // Cooperative load of the 32x128 A stripe (1024 float4s across 256 threads)
    for (int i = tid; i < 1024; i += 256) {
        int r  = i >> 5;                  // 0..31
        int c4 = i & 31;                  // float4 index within row
        int row = row0 + r;
        if (row >= N) row = N - 1;        // duplicated load; never used in epilogue
        float4 v = ((const float4*)(A + (size_t)row * DDIM))[c4];
        float* p = &As[r * LDS_STRIDE + c4 * 4];
        p[0] = v.x; p[1] = v.y; p[2] = v.z; p[3] = v.w;
    }
    __syncthreads();

    const int n0 = wave * 16;
    const int nl = lane & 15;
    const int m  = lane & 15;
    const int kb = (lane >> 4) * 2;

    v8f c0 = {0.f, 0.f, 0.f, 0.f, 0.f, 0.f, 0.f, 0.f};
    v8f c1 = {0.f, 0.f, 0.f, 0.f, 0.f, 0.f, 0.f, 0.f};

    #pragma unroll 4
    for (int k0 = 0; k0 < DDIM; k0 += 4) {
        v2f b;
        b.x = W[(size_t)(k0 + kb)     * DDIM + n0 + nl];
        b.y = W[(size_t)(k0 + kb + 1) * DDIM + n0 + nl];
        v2f a0, a1;
        a0.x = As[m * LDS_STRIDE + k0 + kb];
        a0.y = As[m * LDS_STRIDE + k0 + kb + 1];
        a1.x = As[(m + 16) * LDS_STRIDE + k0 + kb];
        a1.y = As[(m + 16) * LDS_STRIDE + k0 + kb + 1];
        c0 = __builtin_amdgcn_wmma_f32_16x16x4_f32(
                 false, a0, false, b, (short)0, c0, false, false);
        c1 = __builtin_amdgcn_wmma_f32_16x16x4_f32(
                 false, a1, false, b, (short)0, c1, false, false);
    }

    const float bv    = bias[n0 + nl];
    const int   mbase = (lane >> 4) * 8;
    float* op = out + (size_t)(row0 + mbase) * DDIM + n0 + nl;

    if (row0 + 32 <= N) {
        // Fast path: block fully in range, no per-element exec-mask branching.
        #pragma unroll
        for (int i = 0; i < 8; ++i) {
            float v = c0[i] + bv;
            v = v > 0.f ? v : 0.f;
            op[(size_t)i * DDIM] += v;          // unique (row,col) per lane
        }
        #pragma unroll
        for (int i = 0; i < 8; ++i) {
            float v = c1[i] + bv;
            v = v > 0.f ? v : 0.f;
            op[(size_t)(i + 16) * DDIM] += v;
        }
    } else {
        // Ragged tail block (N % 32 != 0): guard each row.
        #pragma unroll
        for (int i = 0; i < 8; ++i) {
            if (row0 + mbase + i < N) {
                float v = c0[i] + bv;
                v = v > 0.f ? v : 0.f;
                op[(size_t)i * DDIM] += v;
            }
        }
        #pragma unroll
        for (int i = 0; i < 8; ++i) {
            if (row0 + mbase + i + 16 < N) {
                float v = c1[i] + bv;
                v = v > 0.f ? v : 0.f;
                op[(size_t)(i + 16) * DDIM] += v;
            }
        }
    }
}

// -------------------------------------------------------------------------
// x_next = x + relu(acc), float4 vectorized
// -------------------------------------------------------------------------
__global__ void __launch_bounds__(256)
residual_relu_kernel(const float* __restrict__ xin,
                     const float* __restrict__ acc,
                     float* __restrict__ xout,
                     int n4)
{
    int i = blockIdx.x * blockDim.x + threadIdx.x;
    if (i >= n4) return;
    float4 xv = ((const float4*)xin)[i];
    float4 av = ((const float4*)acc)[i];
    float4 o;
    o.x = xv.x + fmaxf(av.x, 0.f);
    o.y = xv.y + fmaxf(av.y, 0.f);
    o.z = xv.z + fmaxf(av.z, 0.f);
    o.w = xv.w + fmaxf(av.w, 0.f);
    ((float4*)xout)[i] = o;
}

// -------------------------------------------------------------------------
// Orchestration. Inputs (setup_inputs order):
//   0: x       [N,128] f32          1: Ws_s [3,128,128] f32
//   2: bs_s    [3,128] f32          3: Ws_k [3,3,128,128] f32
//   4: bs_k    [3,3,128] f32        5: edge_index [2,E] i32
//   6: edge_attr [E] i32
// Output: x after 3 layers, [N,128] f32.
// Workspace: 4 * N*D floats (x1, x2, agg, acc) ~= 102 MB.
// -------------------------------------------------------------------------
extern "C" void kernel_launch(void* const* d_in, const int* in_sizes, int n_in,
                              void* d_out, int out_size, void* d_ws, size_t ws_size,
                              hipStream_t stream)
{
    const float* x    = (const float*)d_in[0];
    const float* Ws_s = (const float*)d_in[1];
    const float* bs_s = (const float*)d_in[2];
    const float* Ws_k = (const float*)d_in[3];
    const float* bs_k = (const float*)d_in[4];
    const int*   ei   = (const int*)d_in[5];
    const int*   attr = (const int*)d_in[6];

    const int L = 3;
    const int N = in_sizes[0] / DDIM;
    const int E = in_sizes[6];
    const int* srcI = ei;        // edge_index[0]
    const int* dstI = ei + E;    // edge_index[1]

    const size_t ND = (size_t)N * DDIM;

    float* x1  = (float*)d_ws;
    float* x2  = x1 + ND;
    float* agg = x2 + ND;
    float* acc = agg + ND;

    const float* hist[4];
    hist[0] = x;
    float* outs[3] = { x1, x2, (float*)d_out };

    const int gemmGrid = (N + 31) / 32;
    const int scatGrid = (E + 7) / 8;          // 8 waves (edges) per block
    const int resGrid  = (int)((ND / 4 + 255) / 256);

    const float* xt = x;
    for (int t = 0; t < L; ++t) {
        hipMemsetAsync(acc, 0, ND * sizeof(float), stream);

        for (int k = 1; k <= t + 1; ++k) {
            hipMemsetAsync(agg, 0, ND * sizeof(float), stream);
            const float* h = hist[t - (k - 1)];     // delayed history xs[t-(k-1)]
            scatter_add_kernel<<<scatGrid, 256, 0, stream>>>(
                h, srcI, dstI, attr, k - 1, agg, E);

            const float* Wk = Ws_k + ((size_t)t * L + (k - 1)) * DDIM * DDIM;
            const float* bk = bs_k + ((size_t)t * L + (k - 1)) * DDIM;
            gemm32_relu_acc_kernel<<<gemmGrid, 256, 0, stream>>>(agg, Wk, bk, acc, N);
        }

        // self term: relu(x_t @ Ws_s[t] + bs_s[t])  (eps = 0)
        gemm32_relu_acc_kernel<<<gemmGrid, 256, 0, stream>>>(
            xt, Ws_s + (size_t)t * DDIM * DDIM, bs_s + (size_t)t * DDIM, acc, N);

        // x_{t+1} = x_t + relu(acc)
        residual_relu_kernel<<<resGrid, 256, 0, stream>>>(
            xt, acc, outs[t], (int)(ND / 4));

        hist[t + 1] = outs[t];
        xt = outs[t];
    }
}